// waspGridSpatialIntegral_5669356834488
// MI455X (gfx1250) — compile-verified
//
#include <hip/hip_runtime.h>
#include <hip/hip_bf16.h>

#define W 512
#define NB 64

typedef float v2f __attribute__((ext_vector_type(2)));
typedef float v4f __attribute__((ext_vector_type(4)));
typedef float v8f __attribute__((ext_vector_type(8)));

// ---------------------------------------------------------------------------
// Channel 1: inclusive cumsum along y (axis -2) via fp32 WMMA.
// For each 16(row)x16(col) tile X:  D = L * X + C
//   L = 16x16 lower-triangular ones (constant, lives in A operand)
//   X = tile (B operand), C = per-column carry broadcast down all 16 rows.
// Implemented as 4 chained V_WMMA_F32_16X16X4_F32 (K = 4 each).
// One wave per (batch, 16-column strip); 32 sequential row-tiles with carry.
// Output is written once and never re-read -> non-temporal stores.
// ---------------------------------------------------------------------------
__global__ __launch_bounds__(32) void wasp_cumsum_y_wmma(
    const float* __restrict__ in, float* __restrict__ out) {
  const int lane = threadIdx.x;      // wave32: 0..31
  const int n    = lane & 15;        // A: row M; B/C/D: column N
  const int half = lane >> 4;        // 0 or 1

  const int tile = blockIdx.x & 31;  // 32 column strips of width 16
  const int b    = blockIdx.x >> 5;  // batch

  const size_t chan1 = (size_t)b * 2u * W * W + (size_t)W * W;
  const float* src = in  + chan1 + (size_t)tile * 16 + n;
  float*       dst = out + chan1 + (size_t)tile * 16 + n;

  // Constant A fragments: A[m][k] = (k <= m) ? 1 : 0, with m = lane%16,
  // k = 4*kb + 2*half + v  (f32 16x4 A layout: half-wave holds upper K pair).
  v2f a[4];
#pragma unroll
  for (int kb = 0; kb < 4; ++kb) {
    const int k0 = 4 * kb + 2 * half;
    a[kb].x = (k0 + 0 <= n) ? 1.0f : 0.0f;
    a[kb].y = (k0 + 1 <= n) ? 1.0f : 0.0f;
  }

  float carry = 0.0f;  // running column sum (per lane's column)

  for (int t = 0; t < W / 16; ++t) {
    const float* rp = src + (size_t)t * 16 * W;

    // B fragments: lane l, vgpr v holds X[k = 4*kb + 2*half + v][n].
    // Each scalar load: lanes 0-15 read 16 consecutive floats (64B segment),
    // lanes 16-31 read another 64B segment two rows down.
    v2f bm[4];
#pragma unroll
    for (int kb = 0; kb < 4; ++kb) {
      const int k0 = 4 * kb + 2 * half;
      bm[kb].x = rp[(size_t)(k0 + 0) * W];
      bm[kb].y = rp[(size_t)(k0 + 1) * W];
    }

    // Accumulator seeded with the carry broadcast down the column.
    v8f d;
#pragma unroll
    for (int i = 0; i < 8; ++i) d[i] = carry;

#pragma unroll
    for (int kb = 0; kb < 4; ++kb) {
      d = __builtin_amdgcn_wmma_f32_16x16x4_f32(
          /*neg_a=*/false, a[kb], /*neg_b=*/false, bm[kb],
          /*c_mod=*/(short)0, d, /*reuse_a=*/false, /*reuse_b=*/false);
    }

    // Store D: vgpr v holds out[row = v + 8*half][col]; 64B segments.
    // Non-temporal: output stream has zero reuse, keep it out of L2's way.
    float* wp = dst + (size_t)t * 16 * W;
#pragma unroll
    for (int v = 0; v < 8; ++v) {
      __builtin_nontemporal_store(d[v], wp + (size_t)(v + 8 * half) * W);
    }

    // New carry = D[15, n], held in d[7] of lane 16 + n.
    carry = __shfl(d[7], 16 + n, 32);
  }
}

// ---------------------------------------------------------------------------
// Channel 0: inclusive cumsum along x (contiguous axis).
// One thread per image row; float4 loads/stores (16B aligned), serial scan.
// Cachelines are fully consumed across the loop -> compulsory traffic only.
// Loads keep default temporal hint (short-lived L0 line reuse across the
// 8 float4 accesses per 128B line); stores are non-temporal (written once).
// ---------------------------------------------------------------------------
__global__ __launch_bounds__(256) void wasp_cumsum_x(
    const float* __restrict__ in, float* __restrict__ out) {
  const int r = blockIdx.x * blockDim.x + threadIdx.x;  // 0 .. NB*W-1
  const int b = r >> 9;       // r / 512
  const int y = r & (W - 1);  // r % 512

  const size_t base = (size_t)b * 2u * W * W + (size_t)y * W;  // channel 0
  const v4f* src = (const v4f*)(in + base);
  v4f*       dst = (v4f*)(out + base);

  float acc = 0.0f;
#pragma unroll 4
  for (int i = 0; i < W / 4; ++i) {
    v4f v = src[i];
    v4f o;
    o.x = acc + v.x;
    o.y = o.x + v.y;
    o.z = o.y + v.z;
    o.w = o.z + v.w;
    acc = o.w;
    __builtin_nontemporal_store(o, dst + i);
  }
}

extern "C" void kernel_launch(void* const* d_in, const int* in_sizes, int n_in,
                              void* d_out, int out_size, void* d_ws, size_t ws_size,
                              hipStream_t stream) {
  (void)in_sizes; (void)n_in; (void)d_ws; (void)ws_size; (void)out_size;
  const float* in  = (const float*)d_in[0];
  float*       out = (float*)d_out;

  // Channel 0: x-integral. 64*512 rows, one thread each.
  wasp_cumsum_x<<<(NB * W) / 256, 256, 0, stream>>>(in, out);

  // Channel 1: y-integral via WMMA. One wave per (batch, 16-col strip).
  wasp_cumsum_y_wmma<<<NB * (W / 16), 32, 0, stream>>>(in, out);
}